// MolecularPropertyPrediction_14422500180655
// MI455X (gfx1250) — compile-verified
//
#include <hip/hip_runtime.h>

typedef __attribute__((ext_vector_type(2))) float v2f;
typedef __attribute__((ext_vector_type(8))) float v8f;

#define DIM    256
#define ATOMS  32
#define NMOL   512
#define DTOT   (NMOL * ATOMS)
#define LAYERS 3

// One workgroup (8 wave32 waves) per molecule, fully fused:
// embed gather -> 3x (GEMM+relu, block-adjacency GEMM + residual) -> pool -> head.
// Each wave owns a 2x2 grid of 16x16 output tiles (register blocking: 2 A-frags +
// 2 B-frags -> 4 WMMAs, halving global B-fragment traffic vs a 1x4 tile).
__launch_bounds__(256)
__global__ void molprop_fused_kernel(
    const int*   __restrict__ atoms,
    const float* __restrict__ adjacency,
    const float* __restrict__ embed,
    const float* __restrict__ W_atom,
    const float* __restrict__ b_atom,
    const float* __restrict__ W_prop,
    const float* __restrict__ b_prop,
    float*       __restrict__ out)
{
    __shared__ float lds_x[ATOMS * DIM];  // 32 KB: current features x
    __shared__ float lds_h[ATOMS * DIM];  // 32 KB: relu(xW+b); reused as reduce scratch

    const int mol   = blockIdx.x;
    const int tid   = threadIdx.x;
    const int lane  = tid & 31;
    const int wave  = tid >> 5;
    const int abase = mol * ATOMS;

    // ---- embedding gather: x[a][:] = embed[atoms[abase+a]][:] (float4 vectorized)
    for (int q = tid; q < (ATOMS * DIM) / 4; q += 256) {
        const int a   = q >> 6;          // 64 float4 per 256-wide row
        const int c4  = q & 63;
        const int row = atoms[abase + a];
        ((float4*)lds_x)[q] = ((const float4*)embed)[row * (DIM / 4) + c4];
    }
    __syncthreads();

    // Tile ownership: wave w -> m-tiles {0,1} x n-tiles {2w, 2w+1}  (8 waves = 16 n-tiles)
    const int nta   = wave * 2;
    const int ntb   = wave * 2 + 1;
    const int lhalf = lane >> 4;         // 0: K0/K1 (or M rows +0), 1: K2/K3 (or M rows +8)
    const int lmod  = lane & 15;
    const int koff  = lhalf * 2;         // fragment K offset per ISA 32-bit A layout
    const int na    = nta * 16 + lmod;   // B columns / C columns this lane touches
    const int nb    = ntb * 16 + lmod;

    for (int layer = 0; layer < LAYERS; ++layer) {
        const float* __restrict__ W  = W_atom + layer * DIM * DIM;
        const float* __restrict__ bb = b_atom + layer * DIM;

        // ---- pass 1: hs = relu(x @ W + b); 2x2 tiles: c[m][n]
        v8f c00, c01, c10, c11;
        {
            const float ba = bb[na];
            const float bbv = bb[nb];
#pragma unroll
            for (int v = 0; v < 8; ++v) { c00[v] = ba; c01[v] = bbv; c10[v] = ba; c11[v] = bbv; }
        }
#pragma unroll 4
        for (int k0 = 0; k0 < DIM; k0 += 4) {
            // A (16x4 fp32) for both m-tiles: lanes 0-15 hold K0/K1, lanes 16-31 K2/K3
            const v2f a0 = *(const v2f*)&lds_x[(lmod)      * DIM + k0 + koff];
            const v2f a1 = *(const v2f*)&lds_x[(16 + lmod) * DIM + k0 + koff];
            // B (4x16 fp32) for both n-tiles: rows k0+koff, k0+koff+1 of L2-resident W
            const float* Wk = W + (k0 + koff) * DIM;
            v2f b0, b1;
            b0.x = Wk[na]; b0.y = Wk[DIM + na];
            b1.x = Wk[nb]; b1.y = Wk[DIM + nb];
            c00 = __builtin_amdgcn_wmma_f32_16x16x4_f32(false, a0, false, b0, (short)0, c00, false, false);
            c01 = __builtin_amdgcn_wmma_f32_16x16x4_f32(false, a0, false, b1, (short)0, c01, false, false);
            c10 = __builtin_amdgcn_wmma_f32_16x16x4_f32(false, a1, false, b0, (short)0, c10, false, false);
            c11 = __builtin_amdgcn_wmma_f32_16x16x4_f32(false, a1, false, b1, (short)0, c11, false, false);
        }
        // relu + store hs (C/D layout: VGPR v -> M = mt*16 + v + 8*lhalf, N = nt*16 + lmod)
#pragma unroll
        for (int v = 0; v < 8; ++v) {
            const int r0 = (v + 8 * lhalf) * DIM;        // m-tile 0
            const int r1 = (16 + v + 8 * lhalf) * DIM;   // m-tile 1
            lds_h[r0 + na] = fmaxf(c00[v], 0.0f);
            lds_h[r0 + nb] = fmaxf(c01[v], 0.0f);
            lds_h[r1 + na] = fmaxf(c10[v], 0.0f);
            lds_h[r1 + nb] = fmaxf(c11[v], 0.0f);
        }
        __syncthreads();

        // ---- pass 2: x = x + A_blk(32x32) @ hs ; residual folded into accumulator init
#pragma unroll
        for (int v = 0; v < 8; ++v) {
            const int r0 = (v + 8 * lhalf) * DIM;
            const int r1 = (16 + v + 8 * lhalf) * DIM;
            c00[v] = lds_x[r0 + na];
            c01[v] = lds_x[r0 + nb];
            c10[v] = lds_x[r1 + na];
            c11[v] = lds_x[r1 + nb];
        }
        // adjacency A-fragments: aligned 8-byte (v2f) loads from the L2-hot diagonal block
        const float* Arow0 = adjacency + (long)(abase + lmod)      * DTOT + abase;
        const float* Arow1 = adjacency + (long)(abase + 16 + lmod) * DTOT + abase;
#pragma unroll
        for (int k0 = 0; k0 < ATOMS; k0 += 4) {
            const v2f a0 = *(const v2f*)&Arow0[k0 + koff];
            const v2f a1 = *(const v2f*)&Arow1[k0 + koff];
            const float* Hk = &lds_h[(k0 + koff) * DIM];
            v2f b0, b1;
            b0.x = Hk[na]; b0.y = Hk[DIM + na];
            b1.x = Hk[nb]; b1.y = Hk[DIM + nb];
            c00 = __builtin_amdgcn_wmma_f32_16x16x4_f32(false, a0, false, b0, (short)0, c00, false, false);
            c01 = __builtin_amdgcn_wmma_f32_16x16x4_f32(false, a0, false, b1, (short)0, c01, false, false);
            c10 = __builtin_amdgcn_wmma_f32_16x16x4_f32(false, a1, false, b0, (short)0, c10, false, false);
            c11 = __builtin_amdgcn_wmma_f32_16x16x4_f32(false, a1, false, b1, (short)0, c11, false, false);
        }
        // write updated x back (tiles are wave-disjoint)
#pragma unroll
        for (int v = 0; v < 8; ++v) {
            const int r0 = (v + 8 * lhalf) * DIM;
            const int r1 = (16 + v + 8 * lhalf) * DIM;
            lds_x[r0 + na] = c00[v];
            lds_x[r0 + nb] = c01[v];
            lds_x[r1 + na] = c10[v];
            lds_x[r1 + nb] = c11[v];
        }
        __syncthreads();
    }

    // ---- pooling (sum over 32 atoms) + 256->2 head, deterministic tree reduction
    float y = 0.0f;
#pragma unroll
    for (int a = 0; a < ATOMS; ++a) y += lds_x[a * DIM + tid];

    float* red0 = lds_h;         // hs is dead now; reuse as scratch
    float* red1 = lds_h + 256;
    red0[tid] = y * W_prop[tid * 2 + 0];
    red1[tid] = y * W_prop[tid * 2 + 1];
    __syncthreads();
    for (int s = 128; s > 0; s >>= 1) {
        if (tid < s) {
            red0[tid] += red0[tid + s];
            red1[tid] += red1[tid + s];
        }
        __syncthreads();
    }
    if (tid == 0) {
        out[mol * 2 + 0] = red0[0] + b_prop[0];
        out[mol * 2 + 1] = red1[0] + b_prop[1];
    }
}

extern "C" void kernel_launch(void* const* d_in, const int* in_sizes, int n_in,
                              void* d_out, int out_size, void* d_ws, size_t ws_size,
                              hipStream_t stream) {
    const int*   atoms_p   = (const int*)  d_in[0];
    const float* adjacency = (const float*)d_in[1];
    const float* embed     = (const float*)d_in[2];
    const float* W_atom    = (const float*)d_in[3];
    const float* b_atom    = (const float*)d_in[4];
    const float* W_prop    = (const float*)d_in[5];
    const float* b_prop    = (const float*)d_in[6];
    float* out = (float*)d_out;

    molprop_fused_kernel<<<NMOL, 256, 0, stream>>>(
        atoms_p, adjacency, embed, W_atom, b_atom, W_prop, b_prop, out);
}